// KANLayer_87428354277495
// MI455X (gfx1250) — compile-verified
//
#include <hip/hip_runtime.h>

typedef float v2f __attribute__((ext_vector_type(2)));
typedef float v8f __attribute__((ext_vector_type(8)));

#define IN_DIM  128
#define OUT_DIM 128
#define BATCH   1024
#define NB      6      // n_basis
#define NG      10     // grid knots per edge

// ---------------------------------------------------------------------------
// Kernel 0: zero the spl_reg region of d_out (accumulated via atomics).
// ---------------------------------------------------------------------------
__global__ __launch_bounds__(256)
void kan_zero_sreg(float* __restrict__ sreg, int n) {
    int t = blockIdx.x * blockDim.x + threadIdx.x;
    if (t < n) sreg[t] = 0.0f;
}

// ---------------------------------------------------------------------------
// Kernel 2 (epilogue): sreg[e] *= 1 / (grid[e,9] - grid[e,0] + 1e-5)
// Keeps the f32 divide sequence OUT of the hot WMMA loop.
// ---------------------------------------------------------------------------
__global__ __launch_bounds__(256)
void kan_scale_sreg(float* __restrict__ sreg, const float* __restrict__ grid, int n) {
    int e = blockIdx.x * blockDim.x + threadIdx.x;
    if (e < n) {
        float g0 = grid[e * NG + 0];
        float g9 = grid[e * NG + NG - 1];
        sreg[e] = sreg[e] / (g9 - g0 + 1e-5f);
    }
}

// ---------------------------------------------------------------------------
// Main kernel. Block = 128 threads = 4 wave32.
//   blockIdx.x : batch tile  [16*bx, 16*bx+16)
//   blockIdx.y : output half [64*by, 64*by+64); wave w -> o-tile 64*by + 16*w
// Phase 1: thread t owns column i=t; builds reciprocal tables once (24 divs),
//          then evaluates 16 batch rows with pure FMAs into LDS
//          layout [i][b_local][8] = {B0..B5, swish, 0}.
// Phase 2: per-i dual WMMA chains (V_WMMA_F32_16X16X4_F32):
//   Y chain: accumulates across i; B = {cb0..cb5}*c_spl, K-slot 6 = c_res
//   S chain: raw spl tile per i -> |spl| reduction -> atomicAdd (unscaled)
// ---------------------------------------------------------------------------
__global__ __launch_bounds__(128)
void kan_main(const float* __restrict__ x,
              const float* __restrict__ grid,
              const float* __restrict__ c_basis,
              const float* __restrict__ c_spl,
              const float* __restrict__ c_res,
              float* __restrict__ y,
              float* __restrict__ sreg) {
    __shared__ float basis8[IN_DIM][16][8];   // 64 KB

    const int tid   = threadIdx.x;            // 0..127
    const int bbase = blockIdx.x * 16;

    // ---------------- Phase 1: basis + swish into LDS ----------------------
    {
        const int i = tid;                    // one input column per thread
        float g[NG];
        #pragma unroll
        for (int t = 0; t < NG; ++t) g[t] = grid[i * NG + t];

        // Reciprocal tables: rK[t] = 1/(g[t+K]-g[t]).
        // Note right-denominator(K,t) == left-denominator(K,t+1).
        float r1[9], r2[8], r3[7];
        #pragma unroll
        for (int t = 0; t < 9; ++t) r1[t] = 1.0f / (g[t + 1] - g[t]);
        #pragma unroll
        for (int t = 0; t < 8; ++t) r2[t] = 1.0f / (g[t + 2] - g[t]);
        #pragma unroll
        for (int t = 0; t < 7; ++t) r3[t] = 1.0f / (g[t + 3] - g[t]);

        for (int bl = 0; bl < 16; ++bl) {
            float xv = x[(bbase + bl) * IN_DIM + i];
            float bb[9];
            #pragma unroll
            for (int t = 0; t < 9; ++t)
                bb[t] = (xv >= g[t] && xv < g[t + 1]) ? 1.0f : 0.0f;
            #pragma unroll
            for (int t = 0; t < 8; ++t)
                bb[t] = (xv - g[t]) * r1[t] * bb[t] + (g[t + 2] - xv) * r1[t + 1] * bb[t + 1];
            #pragma unroll
            for (int t = 0; t < 7; ++t)
                bb[t] = (xv - g[t]) * r2[t] * bb[t] + (g[t + 3] - xv) * r2[t + 1] * bb[t + 1];
            #pragma unroll
            for (int t = 0; t < 6; ++t)
                bb[t] = (xv - g[t]) * r3[t] * bb[t] + (g[t + 4] - xv) * r3[t + 1] * bb[t + 1];

            #pragma unroll
            for (int j = 0; j < NB; ++j) basis8[i][bl][j] = bb[j];
            basis8[i][bl][6] = xv / (1.0f + __expf(-xv));   // swish
            basis8[i][bl][7] = 0.0f;                        // zero K-slot
        }
    }
    __syncthreads();

    // ---------------- Phase 2: WMMA over i ---------------------------------
    const int lane = tid & 31;
    const int wave = tid >> 5;               // 0..3
    const int hsel = lane >> 4;              // 0: K rows {0,1}; 1: {2,3}
    const int nn   = lane & 15;              // M (A) / N (B,C,D)
    const int o    = blockIdx.y * 64 + wave * 16 + nn;
    const int K0   = hsel ? 2 : 0;

    v8f cy = {0.f, 0.f, 0.f, 0.f, 0.f, 0.f, 0.f, 0.f};

    for (int i = 0; i < IN_DIM; ++i) {
        // A operands (16x4 f32: VGPR0 = {K0|K2}, VGPR1 = {K1|K3})
        v2f a0 = { basis8[i][nn][K0],     basis8[i][nn][K0 + 1] };
        v2f a1 = { basis8[i][nn][K0 + 4], basis8[i][nn][K0 + 5] };
        // hsel==0 -> {B4,B5};  hsel==1 -> {swish, 0}

        const int    e   = o * IN_DIM + i;
        const float* cb  = c_basis + e * NB;
        const float  csv = c_spl[e];
        const float  crv = c_res[e];

        float cbA = cb[K0], cbB = cb[K0 + 1];
        float cb4 = cb[4],  cb5 = cb[5];

        v2f sb0 = { cbA, cbB };
        v2f sb1 = { hsel ? 0.0f : cb4, hsel ? 0.0f : cb5 };
        v2f yb0 = { cbA * csv, cbB * csv };
        v2f yb1 = { hsel ? crv : cb4 * csv, hsel ? 0.0f : cb5 * csv };

        cy = __builtin_amdgcn_wmma_f32_16x16x4_f32(false, a0, false, yb0,
                                                   (short)0, cy, false, false);
        cy = __builtin_amdgcn_wmma_f32_16x16x4_f32(false, a1, false, yb1,
                                                   (short)0, cy, false, false);

        v8f s8 = {0.f, 0.f, 0.f, 0.f, 0.f, 0.f, 0.f, 0.f};
        s8 = __builtin_amdgcn_wmma_f32_16x16x4_f32(false, a0, false, sb0,
                                                   (short)0, s8, false, false);
        s8 = __builtin_amdgcn_wmma_f32_16x16x4_f32(false, a1, false, sb1,
                                                   (short)0, s8, false, false);

        float part = 0.0f;
        #pragma unroll
        for (int r = 0; r < 8; ++r) part += fabsf(s8[r]);
        part += __shfl_xor(part, 16, 32);     // combine M rows 0..7 with 8..15
        if (lane < 16)
            atomicAdd(&sreg[e], part * (1.0f / (float)BATCH));  // unscaled here
    }

    // ---------------- write y tile -----------------------------------------
    #pragma unroll
    for (int r = 0; r < 8; ++r) {
        int b = bbase + r + 8 * hsel;         // lanes 16-31 hold M = r+8
        y[b * OUT_DIM + o] = cy[r] * (1.0f / (float)IN_DIM);
    }
}

// ---------------------------------------------------------------------------
extern "C" void kernel_launch(void* const* d_in, const int* in_sizes, int n_in,
                              void* d_out, int out_size, void* d_ws, size_t ws_size,
                              hipStream_t stream) {
    (void)in_sizes; (void)n_in; (void)d_ws; (void)ws_size; (void)out_size;

    const float* x       = (const float*)d_in[0];
    const float* grid    = (const float*)d_in[1];
    const float* c_basis = (const float*)d_in[2];
    const float* c_spl   = (const float*)d_in[3];
    const float* c_res   = (const float*)d_in[4];

    float* out  = (float*)d_out;
    float* y    = out;                      // (1024, 128)
    float* sreg = out + BATCH * OUT_DIM;    // (128, 128)

    const int nreg = OUT_DIM * IN_DIM;
    kan_zero_sreg<<<(nreg + 255) / 256, 256, 0, stream>>>(sreg, nreg);

    dim3 gridDim(BATCH / 16, 2);            // 128 blocks
    kan_main<<<gridDim, 128, 0, stream>>>(x, grid, c_basis, c_spl, c_res, y, sreg);

    kan_scale_sreg<<<(nreg + 255) / 256, 256, 0, stream>>>(sreg, grid, nreg);
}